// TotalLoss_15607911154195
// MI455X (gfx1250) — compile-verified
//
#include <hip/hip_runtime.h>
#include <hip/hip_bf16.h>
#include <math.h>

// ---------------- problem constants (fixed by the reference) ----------------
#define BSZ   32
#define NPTS  131072
#define IMG_H 375
#define IMG_W 1242
#define DEPTH_ELEMS ((size_t)BSZ * IMG_H * IMG_W)   // 14,904,000 floats

// workspace layout (floats)
//  [0..31]   per-batch point-cloud error sums
//  [32]      depth derr accumulator (sum over b, cols of sqrt(col SSE))
//  [33]      translation SSE sum, [34] rotation SSE sum
//  [64..575] predicted RT matrices (32 x 16)
//  [576.. ]  depth scatter buffer (BSZ*H*W floats, ~59.6 MB)
#define WS_ERR    0
#define WS_DACC   32
#define WS_LT     33
#define WS_LR     34
#define WS_RT     64
#define WS_DEPTH  576

typedef float v2f __attribute__((ext_vector_type(2)));
typedef float v4f __attribute__((ext_vector_type(4)));
typedef float v8f __attribute__((ext_vector_type(8)));

// ---------------------------------------------------------------------------
// Kernel 0: zero the depth scratch buffer + accumulator header (every call).
// ---------------------------------------------------------------------------
__global__ __launch_bounds__(256) void tl_zero(float4* __restrict__ depth4,
                                               int n4, float* __restrict__ hdr) {
  int i = blockIdx.x * 256 + threadIdx.x;
  if (i < n4) depth4[i] = make_float4(0.f, 0.f, 0.f, 0.f);
  if (blockIdx.x == 0 && threadIdx.x < 64) hdr[threadIdx.x] = 0.f;
}

// ---------------------------------------------------------------------------
// Kernel 1: one lane per batch -- Rodrigues for predicted pose -> pred RT,
// plus translation/rotation SSE sums (wave32 reduction).
// ---------------------------------------------------------------------------
__global__ __launch_bounds__(32) void tl_setup(const float* __restrict__ gt_t,
                                               const float* __restrict__ gt_r,
                                               const float* __restrict__ pr_t,
                                               const float* __restrict__ pr_r,
                                               float* __restrict__ ws) {
  int b = threadIdx.x;            // 0..31, exactly one wave
  float wx = pr_r[b * 3 + 0], wy = pr_r[b * 3 + 1], wz = pr_r[b * 3 + 2];
  float th2 = wx * wx + wy * wy + wz * wz;
  float th  = sqrtf(th2);
  float A = sinf(th) / th;
  float Bc = (1.0f - cosf(th)) / th2;

  float kk00 = -(wy * wy + wz * wz);
  float kk11 = -(wx * wx + wz * wz);
  float kk22 = -(wx * wx + wy * wy);
  float kxy = wx * wy, kxz = wx * wz, kyz = wy * wz;

  float* rt = ws + WS_RT + b * 16;
  rt[0]  = 1.0f + Bc * kk00;  rt[1]  = -A * wz + Bc * kxy;
  rt[2]  =  A * wy + Bc * kxz; rt[3]  = pr_t[b * 3 + 0];
  rt[4]  =  A * wz + Bc * kxy; rt[5]  = 1.0f + Bc * kk11;
  rt[6]  = -A * wx + Bc * kyz; rt[7]  = pr_t[b * 3 + 1];
  rt[8]  = -A * wy + Bc * kxz; rt[9]  =  A * wx + Bc * kyz;
  rt[10] = 1.0f + Bc * kk22;   rt[11] = pr_t[b * 3 + 2];
  rt[12] = 0.0f; rt[13] = 0.0f; rt[14] = 0.0f; rt[15] = 1.0f;

  float dt = 0.f, dr = 0.f;
  for (int j = 0; j < 3; ++j) {
    float a = pr_t[b * 3 + j] - gt_t[b * 3 + j]; dt += a * a;
    float c = pr_r[b * 3 + j] - gt_r[b * 3 + j]; dr += c * c;
  }
  for (int off = 16; off; off >>= 1) {
    dt += __shfl_xor(dt, off, 32);
    dr += __shfl_xor(dr, off, 32);
  }
  if (b == 0) { ws[WS_LT] = dt; ws[WS_LR] = dr; }
}

// ---------------------------------------------------------------------------
// Kernel 2: fused point transform (WMMA f32 16x16x4), per-point error,
// projection through K, and depth scatter.
//   grid = (BSZ, 64), block = 256 (8 waves). 512 waves/batch, 16 tiles/wave,
//   16 points/tile -> 131072 points/batch.
// LDS staging is wave-private, so only wave-local DS ordering is needed:
// s_wait_dscnt 0 instead of block barriers.
// ---------------------------------------------------------------------------
#define WAVES_PER_BATCH 512
#define TILES_PER_WAVE  16

__global__ __launch_bounds__(256) void tl_points(const float* __restrict__ pts,
                                                 const float* __restrict__ gt_rt,
                                                 const float* __restrict__ ws_rt,
                                                 const float* __restrict__ kmat,
                                                 float* __restrict__ depth,
                                                 float* __restrict__ errsum) {
  __shared__ float lds[8 * 256];              // 1 KB per wave, private regions

  const int b    = blockIdx.x;
  const int tid  = threadIdx.x;
  const int wave = tid >> 5;
  const int lane = tid & 31;
  const int hf   = lane >> 4;                 // which K half (0: K=0,1  1: K=2,3)
  const int n    = lane & 15;                 // B-matrix column / point-in-tile
  const int waveGlobal = blockIdx.y * 8 + wave;   // 0..511

  // --- build B (4x16): cols 0..3 = gt RT^T, cols 4..7 = pred RT^T, rest 0 ---
  const float* rtg = gt_rt + b * 16;
  const float* rtp = ws_rt + b * 16;
  const int k0 = hf * 2, k1 = hf * 2 + 1;
  v2f Bm; Bm[0] = 0.f; Bm[1] = 0.f;
  if (n < 4)      { Bm[0] = rtg[n * 4 + k0];       Bm[1] = rtg[n * 4 + k1]; }
  else if (n < 8) { Bm[0] = rtp[(n - 4) * 4 + k0]; Bm[1] = rtp[(n - 4) * 4 + k1]; }

  const float fx = kmat[0], cx = kmat[2], fy = kmat[4], cy = kmat[5];
  const float* base = pts + (size_t)b * NPTS * 4;
  float* depthB = depth + (size_t)b * IMG_H * IMG_W;
  float* ldsW = lds + wave * 256;

  float local = 0.f;
  for (int it = 0; it < TILES_PER_WAVE; ++it) {
    const int tile = waveGlobal + it * WAVES_PER_BATCH;
    const int p = tile * 16 + n;                       // point index
    const v4f pt =
        __builtin_nontemporal_load((const v4f*)(base + (size_t)p * 4));
    if (it + 1 < TILES_PER_WAVE)                       // global_prefetch next tile
      __builtin_prefetch(base + (size_t)(p + WAVES_PER_BATCH * 16) * 4, 0, 3);

    // A (16x4 f32): VGPR0 = K0/K2, VGPR1 = K1/K3 per half-wave
    v2f Am;
    Am[0] = hf ? pt[2] : pt[0];
    Am[1] = hf ? pt[3] : pt[1];

    v8f C = {};
    C = __builtin_amdgcn_wmma_f32_16x16x4_f32(
        /*neg_a=*/false, Am, /*neg_b=*/false, Bm,
        /*c_mod=*/(short)0, C, /*reuse_a=*/false, /*reuse_b=*/false);

    // stage D through LDS (wave-private region) to re-lane it point-major
    #pragma unroll
    for (int v = 0; v < 8; ++v) ldsW[v * 32 + lane] = C[v];
    // same-wave LDS ops are in-order; just drain the DS pipe before reading
    asm volatile("s_wait_dscnt 0x0" ::: "memory");

    float e = 0.f;
    if (lane < 16) {
      const int m = lane;                               // point within tile
      const float* q = ldsW + (m & 7) * 32 + (m >> 3) * 16;
      const float4 g  = *(const float4*)(q);            // gt-transformed
      const float4 pr = *(const float4*)(q + 4);        // pred-transformed
      float d0 = pr.x - g.x, d1 = pr.y - g.y, d2 = pr.z - g.z, d3 = pr.w - g.w;
      e = __builtin_amdgcn_sqrtf(d0 * d0 + d1 * d1 + d2 * d2 + d3 * d3);

      const float Z = pr.z;
      const float u = fx * pr.x + cx * Z;
      const float v = fy * pr.y + cy * Z;
      if (Z > 0.f) {
        const float invZ = __builtin_amdgcn_rcpf(Z);    // pixel-grid accuracy
        float xf = fminf(fmaxf(u * invZ, 0.f), (float)(IMG_W - 1));
        float yf = fminf(fmaxf(v * invZ, 0.f), (float)(IMG_H - 1));
        int xi = (int)xf, yi = (int)yf;
        depthB[yi * IMG_W + xi] = Z;                    // scatter (set semantics)
      }
    }
    local += e;
  }

  for (int off = 16; off; off >>= 1) local += __shfl_xor(local, off, 32);
  if (lane == 0) atomicAdd(&errsum[b], local);
}

// ---------------------------------------------------------------------------
// Kernel 3: depth map loss. grid=(5, BSZ), block=256. Each thread owns one
// column; row loop gives fully-coalesced 256-wide loads of depth & gt.
// ---------------------------------------------------------------------------
__global__ __launch_bounds__(256) void tl_depthred(const float* __restrict__ depth,
                                                   const float* __restrict__ gt,
                                                   float* __restrict__ acc) {
  const int b   = blockIdx.y;
  const int col = blockIdx.x * 256 + threadIdx.x;
  float s = 0.f;
  if (col < IMG_W) {
    const float* dp = depth + (size_t)b * IMG_H * IMG_W + col;
    const float* gp = gt    + (size_t)b * IMG_H * IMG_W + col;
    #pragma unroll 5
    for (int r = 0; r < IMG_H; ++r) {
      float d = dp[r * IMG_W] - __builtin_nontemporal_load(gp + r * IMG_W);
      s += d * d;
    }
    s = __builtin_amdgcn_sqrtf(s);      // derr for this (b, col)
  }
  for (int off = 16; off; off >>= 1) s += __shfl_xor(s, off, 32);
  __shared__ float red[8];
  if ((threadIdx.x & 31) == 0) red[threadIdx.x >> 5] = s;
  __syncthreads();
  if (threadIdx.x < 8) {
    float v = red[threadIdx.x];
    for (int off = 4; off; off >>= 1) v += __shfl_xor(v, off, 32);
    if (threadIdx.x == 0) atomicAdd(acc, v);
  }
}

// ---------------------------------------------------------------------------
// Kernel 4: combine accumulators -> (total, trans_loss, depth/B, pc/B)
// ---------------------------------------------------------------------------
__global__ void tl_finalize(const float* __restrict__ ws, float* __restrict__ out) {
  float loss_t = ws[WS_LT] / (float)BSZ;
  float loss_r = ws[WS_LR] / (float)BSZ;
  float transformation_loss = 2.0f * loss_t + 1.0f * loss_r;   // TRANS_W, ROT_W

  float pc_sum = 0.f;
  for (int b = 0; b < BSZ; ++b) pc_sum += ws[WS_ERR + b] / (float)NPTS;
  float pc_term = pc_sum / (float)BSZ;

  float depth_loss = ws[WS_DACC] / (float)IMG_W;               // mean over cols
  float depth_term = depth_loss / (float)BSZ;

  float total = 0.5f * transformation_loss + 1.0f * depth_term + 0.5f * pc_term;
  out[0] = total;
  out[1] = transformation_loss;
  out[2] = depth_term;
  out[3] = pc_term;
}

// ---------------------------------------------------------------------------
extern "C" void kernel_launch(void* const* d_in, const int* in_sizes, int n_in,
                              void* d_out, int out_size, void* d_ws, size_t ws_size,
                              hipStream_t stream) {
  const float* pts      = (const float*)d_in[0];
  const float* gt_t     = (const float*)d_in[1];
  const float* gt_r     = (const float*)d_in[2];
  const float* pr_t     = (const float*)d_in[3];
  const float* pr_r     = (const float*)d_in[4];
  const float* gt_rt    = (const float*)d_in[5];
  const float* kmat     = (const float*)d_in[6];
  const float* gt_depth = (const float*)d_in[7];
  float* out = (float*)d_out;
  float* ws  = (float*)d_ws;
  float* depth = ws + WS_DEPTH;

  const int n4 = (int)(DEPTH_ELEMS / 4);                  // 3,726,000
  tl_zero<<<(n4 + 255) / 256, 256, 0, stream>>>((float4*)depth, n4, ws);
  tl_setup<<<1, 32, 0, stream>>>(gt_t, gt_r, pr_t, pr_r, ws);
  tl_points<<<dim3(BSZ, 64), 256, 0, stream>>>(pts, gt_rt, ws + WS_RT, kmat,
                                               depth, ws + WS_ERR);
  tl_depthred<<<dim3(5, BSZ), 256, 0, stream>>>(depth, gt_depth, ws + WS_DACC);
  tl_finalize<<<1, 1, 0, stream>>>(ws, out);
}